// ActorCritic_attention_76424648065336
// MI455X (gfx1250) — compile-verified
//
#include <hip/hip_runtime.h>

// ---------------------------------------------------------------------------
// Problem constants (from reference)
// ---------------------------------------------------------------------------
#define A_N      8
#define T_N      32768
#define IN_D     29
#define Hd       128
#define KC       16                       // codes per codebook
#define L_TOT    (A_N * T_N)              // 262144
#define WAVES    4
#define ROWS_PER_WG 64                    // 4 waves x 16 rows
#define NBLOCKS  (L_TOT / ROWS_PER_WG)    // 4096
#define LDA      132   // padded row stride (floats) for 16x128 tiles: bank-conflict-free
#define LDAP     36    // padded row stride for the 16x32 proj A-stage

typedef float v8f __attribute__((ext_vector_type(8)));
typedef float v2f __attribute__((ext_vector_type(2)));

// ---------------------------------------------------------------------------
// V_WMMA_F32_16X16X4_F32 : D(16x16 f32) = A(16x4) * B(4x16) + C
// ---------------------------------------------------------------------------
__device__ __forceinline__ v8f wmma4(v2f a, v2f b, v8f c) {
  return __builtin_amdgcn_wmma_f32_16x16x4_f32(false, a, false, b,
                                               (short)0, c, false, false);
}

// Swizzled B layout: element (k,n) lives at ((k>>2)*128 + n)*4 + (k&3).
// A lane's fragment (rows k0+kh, k0+kh+1 at a fixed column) is then one
// aligned 8-byte LDS load, and consecutive lanes touch disjoint banks.
__device__ __forceinline__ int bswz(int k, int n) {
  return (((k >> 2) << 7) + n) * 4 + (k & 3);
}

// Wide GEMM: one 16-row tile x all 128 columns. K-outer: A fragment loaded
// once per K-step and reused for 8 column blocks (8 live accumulators).
__device__ __forceinline__ void gemm_wide(const float* Alds, int lda,
                                          const float* Bsw, int Kdim,
                                          v8f acc[8]) {
  const int lane = threadIdx.x & 31;
  const int row  = lane & 15;
  const int kh   = (lane >> 4) << 1;          // 0 or 2
  for (int k0 = 0; k0 < Kdim; k0 += 4) {
    v2f a = *(const v2f*)(Alds + row * lda + k0 + kh);
#pragma unroll
    for (int j = 0; j < 8; ++j) {
      const int col = (j << 4) + row;
      v2f b = *(const v2f*)(Bsw + (((k0 >> 2) << 7) + col) * 4 + kh);
      acc[j] = wmma4(a, b, acc[j]);
    }
  }
}

// Narrow GEMM (N=16, row-major B): scores / codebooks / head outputs.
__device__ __forceinline__ v8f gemm_n16(const float* Alds, int lda,
                                        const float* Blds, int Kdim, v8f acc) {
  const int lane = threadIdx.x & 31;
  const int row  = lane & 15;
  const int kh   = (lane >> 4) << 1;
  for (int k0 = 0; k0 < Kdim; k0 += 4) {
    v2f a = *(const v2f*)(Alds + row * lda + k0 + kh);
    v2f b;
    b.x = Blds[(k0 + kh) * 16 + row];
    b.y = Blds[(k0 + kh + 1) * 16 + row];
    acc = wmma4(a, b, acc);
  }
  return acc;
}

// C/D tile (ISA 7.12.2): lane<16 -> col=lane, rows v; lane>=16 -> rows v+8.
__device__ __forceinline__ void store_c_lds(float* out, int ldN, v8f c) {
  const int lane  = threadIdx.x & 31;
  const int col   = lane & 15;
  const int rbase = (lane >> 4) << 3;
#pragma unroll
  for (int v = 0; v < 8; ++v) out[(rbase + v) * ldN + col] = c[v];
}

// ---------------------------------------------------------------------------
// Fused forward: one 16-row tile per wave, 4 waves per block.
// ---------------------------------------------------------------------------
__global__ __launch_bounds__(128) void fused_fwd(
    const float* __restrict__ state,  const float* __restrict__ projW,
    const float* __restrict__ projB,  const float* __restrict__ quantK,
    const float* __restrict__ inW,    const float* __restrict__ inB,
    const float* __restrict__ cb0,    const float* __restrict__ cb1,
    const float* __restrict__ cb2,    const float* __restrict__ aW1,
    const float* __restrict__ aB1,    const float* __restrict__ aW2,
    const float* __restrict__ aB2,    const float* __restrict__ cW1,
    const float* __restrict__ cB1,    const float* __restrict__ cW2,
    const float* __restrict__ cB2,    const float* __restrict__ gum,
    float* __restrict__ outAct, float* __restrict__ outVal,
    float* __restrict__ wsErr) {
  __shared__ __align__(16) float wbuf[128 * 128];   // 64KB swizzled B (projW/Wq/aW1/cW1)
  __shared__ __align__(16) float sNB [128 * 16];    // Kmat, later padded W2
  __shared__ __align__(16) float sCbT0[32 * 16];    // cb^T [d][code]
  __shared__ __align__(16) float sCbT1[64 * 16];
  __shared__ __align__(16) float sCbT2[128 * 16];
  __shared__ float sCn0[16], sCn1[16], sCn2[16];
  __shared__ float sProjB[128], sBq[128], sB1[128], sB2[16];
  __shared__ __align__(16) float qbuf[WAVES][16 * LDA];  // q, later ha/hc
  __shared__ __align__(16) float sbuf[WAVES][16 * LDA];  // stage/qq/s
  __shared__ int   sLvl [WAVES][16];
  __shared__ int   sIdx0[WAVES][4][16];
  __shared__ int   sIdx1[WAVES][2][16];
  __shared__ int   sIdx2[WAVES][16];
  __shared__ float sErr[3];

  const int tid   = threadIdx.x;
  const int wave  = tid >> 5;
  const int lane  = tid & 31;
  const int wg    = blockIdx.x;
  const int l0    = wg * ROWS_PER_WG + wave * 16;
  const int agent = (wg * ROWS_PER_WG) / T_N;       // uniform per block

  // Early prefetch (global_prefetch_b8): gumbel rows + MLP weights used later.
  if (lane < 16) __builtin_prefetch(gum + (size_t)(l0 + lane) * 3, 0, 1);
  __builtin_prefetch(aW1 + (size_t)agent * 16384 + tid * 128, 0, 1);
  __builtin_prefetch(cW1 + (size_t)agent * 16384 + tid * 128, 0, 1);

  // -------- stage shared constants --------
  if (tid < 3) sErr[tid] = 0.0f;
  if (tid < 128) { sProjB[tid] = projB[tid]; sBq[tid] = inB[tid]; }
  for (int e = tid; e < KC * 32;  e += 128) { int c = e / 32,  d = e % 32;  sCbT0[d * 16 + c] = cb0[e]; }
  for (int e = tid; e < KC * 64;  e += 128) { int c = e / 64,  d = e % 64;  sCbT1[d * 16 + c] = cb1[e]; }
  for (int e = tid; e < KC * 128; e += 128) { int c = e >> 7,  d = e & 127; sCbT2[d * 16 + c] = cb2[e]; }
  if (tid < 16) {
    float n0 = 0.f, n1 = 0.f, n2 = 0.f;
    for (int d = 0; d < 32;  ++d) { float v = cb0[tid * 32  + d]; n0 += v * v; }
    for (int d = 0; d < 64;  ++d) { float v = cb1[tid * 64  + d]; n1 += v * v; }
    for (int d = 0; d < 128; ++d) { float v = cb2[tid * 128 + d]; n2 += v * v; }
    sCn0[tid] = n0; sCn1[tid] = n1; sCn2[tid] = n2;
  }
  for (int e = tid; e < 128 * 16; e += 128) sNB[e] = 0.0f;   // zero Kmat
  __syncthreads();

  // kk[s][t] = quant_keys[s] . Wk_row[t] + bk[t];  Kmat[32h+d][h*3+s] = kk
  if (tid < 128) {
    float a0 = inB[128 + tid], a1 = a0, a2 = a0;
    const float* wk = inW + (size_t)(128 + tid) * Hd;
    for (int j = 0; j < Hd; ++j) {
      float w = wk[j];
      a0 += quantK[0 * Hd + j] * w;
      a1 += quantK[1 * Hd + j] * w;
      a2 += quantK[2 * Hd + j] * w;
    }
    const int h = tid >> 5;
    sNB[tid * 16 + h * 3 + 0] = a0;
    sNB[tid * 16 + h * 3 + 1] = a1;
    sNB[tid * 16 + h * 3 + 2] = a2;
  }
  // projW (29x128, zero-padded to 32) swizzled into wbuf
  for (int e = tid; e < 32 * 128; e += 128) {
    int k = e >> 7, n = e & 127;
    wbuf[bswz(k, n)] = (k < IN_D) ? projW[e] : 0.0f;
  }
  // stage state tile 16x32 (zero-pad 29->32), stride LDAP, into sbuf[wave]
  for (int e = lane; e < 16 * 32; e += 32) {
    int r = e >> 5, c = e & 31;
    sbuf[wave][r * LDAP + c] = (c < IN_D) ? state[(size_t)(l0 + r) * IN_D + c] : 0.0f;
  }
  __syncthreads();

  const int colL = lane & 15;
  const int rb   = (lane >> 4) << 3;

  // -------- phase 1: x = state @ projW + b  -> qbuf --------
  {
    v8f acc[8] = {};
    gemm_wide(&sbuf[wave][0], LDAP, wbuf, 32, acc);
#pragma unroll
    for (int j = 0; j < 8; ++j) {
      const int col = (j << 4) + colL;
      const float b = sProjB[col];
#pragma unroll
      for (int v = 0; v < 8; ++v) qbuf[wave][(rb + v) * LDA + col] = acc[j][v] + b;
    }
  }
  __syncthreads();

  // -------- phase 2: qq = (q @ Wq^T + bq)/sqrt(32) -> sbuf --------
  for (int e = tid; e < 128 * 128; e += 128) {           // Wq^T, swizzled
    int k = e >> 7, n = e & 127;
    wbuf[bswz(k, n)] = inW[(size_t)n * Hd + k];
  }
  __syncthreads();
  {
    const float isd = 0.17677669529663687f;              // 1/sqrt(32)
    v8f acc[8] = {};
    gemm_wide(&qbuf[wave][0], LDA, wbuf, 128, acc);
#pragma unroll
    for (int j = 0; j < 8; ++j) {
      const int col = (j << 4) + colL;
      const float b = sBq[col];
#pragma unroll
      for (int v = 0; v < 8; ++v)
        sbuf[wave][(rb + v) * LDA + col] = (acc[j][v] + b) * isd;
    }
  }
  __syncthreads();

  // -------- phase 3: scores = qq @ Kmat; gumbel gate --------
  float* scd = &wbuf[wave * 256];   // 16x16 scratch (Wq now dead)
  {
    v8f acc = {};
    acc = gemm_n16(&sbuf[wave][0], LDA, sNB, 128, acc);
    store_c_lds(scd, 16, acc);
  }
  __syncthreads();
  if (lane < 16) {
    const int m = lane;
    float at0 = 0.f, at1 = 0.f, at2 = 0.f;
    for (int h = 0; h < 4; ++h) {
      float s0 = scd[m * 16 + h * 3 + 0];
      float s1 = scd[m * 16 + h * 3 + 1];
      float s2 = scd[m * 16 + h * 3 + 2];
      float mx = fmaxf(s0, fmaxf(s1, s2));
      float e0 = __expf(s0 - mx), e1 = __expf(s1 - mx), e2 = __expf(s2 - mx);
      float inv = 1.0f / (e0 + e1 + e2);
      at0 += e0 * inv; at1 += e1 * inv; at2 += e2 * inv;
    }
    const float* gu = gum + (size_t)(l0 + m) * 3;
    float att[3] = {at0 * 0.25f, at1 * 0.25f, at2 * 0.25f};
    int best = 0; float bv = -3.4e38f;
    for (int s = 0; s < 3; ++s) {
      float g = -__logf(-__logf(gu[s] + 1e-10f) + 1e-10f);
      float y = att[s] + g;                 // softmax monotone: argmax identical
      if (y > bv) { bv = y; best = s; }
    }
    sLvl[wave][m] = best;
  }
  __syncthreads();

  // -------- phase 4: codebook argmin (proxy = |c|^2 - 2 q.c) --------
  {
    v8f acc = {};
    acc = gemm_n16(&qbuf[wave][0], LDA, sCbT2, 128, acc);
    store_c_lds(scd, 16, acc);
  }
  __syncthreads();
  if (lane < 16) {
    int best = 0; float bv = 3.4e38f;
    for (int c = 0; c < KC; ++c) {
      float d = sCn2[c] - 2.0f * scd[lane * 16 + c];
      if (d < bv) { bv = d; best = c; }
    }
    sIdx2[wave][lane] = best;
  }
  __syncthreads();
  for (int half = 0; half < 2; ++half) {
    v8f acc = {};
    acc = gemm_n16(&qbuf[wave][half * 64], LDA, sCbT1, 64, acc);
    store_c_lds(scd, 16, acc);
    __syncthreads();
    if (lane < 16) {
      int best = 0; float bv = 3.4e38f;
      for (int c = 0; c < KC; ++c) {
        float d = sCn1[c] - 2.0f * scd[lane * 16 + c];
        if (d < bv) { bv = d; best = c; }
      }
      sIdx1[wave][half][lane] = best;
    }
    __syncthreads();
  }
  for (int ch = 0; ch < 4; ++ch) {
    v8f acc = {};
    acc = gemm_n16(&qbuf[wave][ch * 32], LDA, sCbT0, 32, acc);
    store_c_lds(scd, 16, acc);
    __syncthreads();
    if (lane < 16) {
      int best = 0; float bv = 3.4e38f;
      for (int c = 0; c < KC; ++c) {
        float d = sCn0[c] - 2.0f * scd[lane * 16 + c];
        if (d < bv) { bv = d; best = c; }
      }
      sIdx0[wave][ch][lane] = best;
    }
    __syncthreads();
  }

  // -------- phase 5: s = recon[gate level]; accumulate CB errors --------
  float e0 = 0.f, e1 = 0.f, e2 = 0.f;
  for (int m = 0; m < 16; ++m) {
    const int lvl = sLvl[wave][m];
    const int i2  = sIdx2[wave][m];
    const int i1a = sIdx1[wave][0][m], i1b = sIdx1[wave][1][m];
    for (int c = lane; c < 128; c += 32) {
      float r2 = sCbT2[c * 16 + i2];
      float r1 = sCbT1[(c & 63) * 16 + ((c < 64) ? i1a : i1b)];
      float r0 = sCbT0[(c & 31) * 16 + sIdx0[wave][c >> 5][m]];
      float qv = qbuf[wave][m * LDA + c];
      float d0 = qv - r0, d1 = qv - r1, d2 = qv - r2;
      e0 += d0 * d0; e1 += d1 * d1; e2 += d2 * d2;
      sbuf[wave][m * LDA + c] = (lvl == 0) ? r0 : ((lvl == 1) ? r1 : r2);
    }
  }
  for (int off = 16; off > 0; off >>= 1) {
    e0 += __shfl_down(e0, off);
    e1 += __shfl_down(e1, off);
    e2 += __shfl_down(e2, off);
  }
  if (lane == 0) {
    atomicAdd(&sErr[0], e0); atomicAdd(&sErr[1], e1); atomicAdd(&sErr[2], e2);
  }
  __syncthreads();
  if (tid < 3) wsErr[(size_t)wg * 3 + tid] = sErr[tid];

  // -------- phase 6: actor MLP --------
  for (int e = tid; e < 128 * 128; e += 128) {
    int k = e >> 7, n = e & 127;
    wbuf[bswz(k, n)] = aW1[(size_t)agent * 16384 + e];
  }
  if (tid < 128) sB1[tid] = aB1[agent * 128 + tid];
  for (int e = tid; e < 128 * 16; e += 128) {            // padded W2 (128x16)
    int k = e >> 4, o = e & 15;
    sNB[e] = (o < 4) ? aW2[(size_t)agent * 512 + k * 4 + o] : 0.0f;
  }
  if (tid < 4) sB2[tid] = aB2[agent * 4 + tid];
  __syncthreads();
  {
    v8f acc[8] = {};
    gemm_wide(&sbuf[wave][0], LDA, wbuf, 128, acc);
#pragma unroll
    for (int j = 0; j < 8; ++j) {
      const int col = (j << 4) + colL;
      const float b = sB1[col];
#pragma unroll
      for (int v = 0; v < 8; ++v)
        qbuf[wave][(rb + v) * LDA + col] = fmaxf(acc[j][v] + b, 0.0f);
    }
  }
  __syncthreads();
  {
    v8f acc = {};
    acc = gemm_n16(&qbuf[wave][0], LDA, sNB, 128, acc);
    if (colL < 4) {
      const float b = sB2[colL];
#pragma unroll
      for (int v = 0; v < 8; ++v)
        outAct[(size_t)(l0 + rb + v) * 4 + colL] = acc[v] + b;
    }
  }
  __syncthreads();

  // -------- phase 7: critic MLP --------
  for (int e = tid; e < 128 * 128; e += 128) {
    int k = e >> 7, n = e & 127;
    wbuf[bswz(k, n)] = cW1[(size_t)agent * 16384 + e];
  }
  if (tid < 128) sB1[tid] = cB1[agent * 128 + tid];
  for (int e = tid; e < 128 * 16; e += 128) {
    int k = e >> 4, o = e & 15;
    sNB[e] = (o == 0) ? cW2[(size_t)agent * 128 + k] : 0.0f;
  }
  if (tid == 0) sB2[0] = cB2[agent];
  __syncthreads();
  {
    v8f acc[8] = {};
    gemm_wide(&sbuf[wave][0], LDA, wbuf, 128, acc);
#pragma unroll
    for (int j = 0; j < 8; ++j) {
      const int col = (j << 4) + colL;
      const float b = sB1[col];
#pragma unroll
      for (int v = 0; v < 8; ++v)
        qbuf[wave][(rb + v) * LDA + col] = fmaxf(acc[j][v] + b, 0.0f);
    }
  }
  __syncthreads();
  {
    v8f acc = {};
    acc = gemm_n16(&qbuf[wave][0], LDA, sNB, 128, acc);
    if (colL == 0) {
      const float b = sB2[0];
#pragma unroll
      for (int v = 0; v < 8; ++v)
        outVal[(size_t)(l0 + rb + v)] = acc[v] + b;
    }
  }
}

// ---------------------------------------------------------------------------
// Deterministic CBloss reduction: sum 4096*3 partials / (3*L*H)
// ---------------------------------------------------------------------------
__global__ __launch_bounds__(256) void cb_reduce(const float* __restrict__ ws,
                                                 float* __restrict__ outCB) {
  __shared__ float red[256];
  float s = 0.0f;
  for (int i = threadIdx.x; i < NBLOCKS * 3; i += 256) s += ws[i];
  red[threadIdx.x] = s;
  __syncthreads();
  for (int o = 128; o > 0; o >>= 1) {
    if (threadIdx.x < o) red[threadIdx.x] += red[threadIdx.x + o];
    __syncthreads();
  }
  if (threadIdx.x == 0)
    outCB[0] = red[0] / (3.0f * (float)L_TOT * (float)Hd);
}

// ---------------------------------------------------------------------------
extern "C" void kernel_launch(void* const* d_in, const int* in_sizes, int n_in,
                              void* d_out, int out_size, void* d_ws,
                              size_t ws_size, hipStream_t stream) {
  (void)in_sizes; (void)n_in; (void)out_size; (void)ws_size;
  const float* state  = (const float*)d_in[0];
  const float* projW  = (const float*)d_in[1];
  const float* projB  = (const float*)d_in[2];
  const float* quantK = (const float*)d_in[3];
  const float* inW    = (const float*)d_in[4];
  const float* inB    = (const float*)d_in[5];
  const float* cb0    = (const float*)d_in[6];
  const float* cb1    = (const float*)d_in[7];
  const float* cb2    = (const float*)d_in[8];
  const float* aW1    = (const float*)d_in[9];
  const float* aB1    = (const float*)d_in[10];
  const float* aW2    = (const float*)d_in[11];
  const float* aB2    = (const float*)d_in[12];
  const float* cW1    = (const float*)d_in[13];
  const float* cB1    = (const float*)d_in[14];
  const float* cW2    = (const float*)d_in[15];
  const float* cB2    = (const float*)d_in[16];
  const float* gum    = (const float*)d_in[17];

  float* out    = (float*)d_out;
  float* outAct = out;                                 // (A,T,4)
  float* outVal = out + (size_t)A_N * T_N * 4;         // (A,T,1)
  float* outCB  = out + (size_t)A_N * T_N * 5;         // scalar
  float* ws     = (float*)d_ws;                        // NBLOCKS*3 partials

  fused_fwd<<<NBLOCKS, 128, 0, stream>>>(
      state, projW, projB, quantK, inW, inB, cb0, cb1, cb2,
      aW1, aB1, aW2, aB2, cW1, cB1, cW2, cB2, gum,
      outAct, outVal, ws);
  cb_reduce<<<1, 256, 0, stream>>>(ws, outCB);
}